// GraphSAGE_FraudDetector_22883585753345
// MI455X (gfx1250) — compile-verified
//
#include <hip/hip_runtime.h>

// GraphSAGE fraud detector for MI455X (gfx1250, wave32).
// Edge aggregation is the bandwidth bottleneck (~1.2 GB @ 23.3 TB/s); GEMMs
// (~10.6 GFLOP) run on exact-precision V_WMMA_F32_16X16X4_F32.

#define NN 50000      // nodes (== 3125 * 16, exact row tiles)
#define NE 800000     // edges
#define DD 128        // feature dim
#define HCOLS 64      // classifier hidden
#define BN_EPS 1e-5f

typedef __attribute__((ext_vector_type(2))) float v2f;
typedef __attribute__((ext_vector_type(8))) float v8f;

// ---------------------------------------------------------------- utilities
__global__ void zero4_kernel(float4* __restrict__ p, int n4) {
  int i = blockIdx.x * blockDim.x + threadIdx.x;
  if (i < n4) p[i] = make_float4(0.f, 0.f, 0.f, 0.f);
}

__global__ void degree_kernel(const int* __restrict__ dst, float* __restrict__ deg) {
  int e = blockIdx.x * blockDim.x + threadIdx.x;
  if (e < NE) atomicAdd(&deg[dst[e]], 1.0f);
}

__global__ void deginv_kernel(float* __restrict__ deg) {
  int i = blockIdx.x * blockDim.x + threadIdx.x;
  if (i < NN) {
    float d = deg[i];
    deg[i] = (d > 0.f) ? (1.0f / d) : 0.f;
  }
}

// 32 lanes per edge, each lane moves 4 floats: coalesced 512B gather per edge,
// scatter-add via global_atomic_add_f32.
__global__ void aggregate_kernel(const float* __restrict__ x, const int* __restrict__ src,
                                 const int* __restrict__ dst, float* __restrict__ agg) {
  int t = blockIdx.x * blockDim.x + threadIdx.x;
  int e = t >> 5;
  if (e >= NE) return;
  int f = (t & 31) << 2;
  const float4 v = *(const float4*)(x + (size_t)src[e] * DD + f);
  float* a = agg + (size_t)dst[e] * DD + f;
  atomicAdd(a + 0, v.x);
  atomicAdd(a + 1, v.y);
  atomicAdd(a + 2, v.z);
  atomicAdd(a + 3, v.w);
}

// ------------------------------------------------------------ WMMA GEMM
// out[16 rows x ncols] = act( A1' @ W1^T (+ A2 @ W2^T) + bias ), A1' = A1 * scale.
// One block per 16-row tile; wave w owns columns [16w, 16w+16).
// K is fixed at 128. Optionally accumulates per-column sum / sum-of-squares
// for training-mode BatchNorm.
template <bool DUAL, bool RELU, bool STATS>
__global__ void wmma_gemm_kernel(const float* A1, const float* __restrict__ scale,
                                 const float* __restrict__ A2,
                                 const float* __restrict__ W1, const float* __restrict__ W2,
                                 const float* __restrict__ bias,
                                 float* out, int ncols,
                                 float* __restrict__ colsum, float* __restrict__ colsumsq) {
  __shared__ float sA1[16][132];  // stride 132: bank = (4*lane + k) % 64, conflict-free
  __shared__ float sA2[16][132];

  const int tid = threadIdx.x;
  const int rbase = blockIdx.x * 16;

  for (int idx = tid; idx < 16 * DD; idx += blockDim.x) {
    const int r = idx >> 7;
    const int k = idx & (DD - 1);
    float v = A1[(size_t)(rbase + r) * DD + k];
    if (scale) v *= scale[rbase + r];
    sA1[r][k] = v;
    if (DUAL) sA2[r][k] = A2[(size_t)(rbase + r) * DD + k];
  }
  __syncthreads();

  const int lane = tid & 31;
  const int wave = tid >> 5;
  const int nloc = lane & 15;            // A row (M) and local output column
  const int kh = (lane >> 4) << 1;       // K sub-offset: lanes 16-31 take K+2,K+3
  const int col = wave * 16 + nloc;

  v8f acc = {};

  const float* w1row = W1 + (size_t)col * DD;  // B[k][n] = W[n][k]
#pragma unroll 4
  for (int kk = 0; kk < DD; kk += 4) {
    const float2 av = *(const float2*)&sA1[nloc][kk + kh];
    const float2 bv = *(const float2*)(w1row + kk + kh);
    v2f a; a.x = av.x; a.y = av.y;
    v2f b; b.x = bv.x; b.y = bv.y;
    acc = __builtin_amdgcn_wmma_f32_16x16x4_f32(false, a, false, b, (short)0, acc,
                                                false, false);
  }
  if (DUAL) {
    const float* w2row = W2 + (size_t)col * DD;
#pragma unroll 4
    for (int kk = 0; kk < DD; kk += 4) {
      const float2 av = *(const float2*)&sA2[nloc][kk + kh];
      const float2 bv = *(const float2*)(w2row + kk + kh);
      v2f a; a.x = av.x; a.y = av.y;
      v2f b; b.x = bv.x; b.y = bv.y;
      acc = __builtin_amdgcn_wmma_f32_16x16x4_f32(false, a, false, b, (short)0, acc,
                                                  false, false);
    }
  }

  const float bz = bias[col];
  const int mbase = rbase + ((lane >> 4) << 3);  // C/D: M = i + 8*(lane>=16)
  float s = 0.f, s2 = 0.f;
#pragma unroll
  for (int i = 0; i < 8; ++i) {
    float v = acc[i] + bz;
    if (RELU) v = fmaxf(v, 0.f);
    out[(size_t)(mbase + i) * ncols + col] = v;
    if (STATS) { s += v; s2 = fmaf(v, v, s2); }
  }
  if (STATS) {
    atomicAdd(&colsum[col], s);
    atomicAdd(&colsumsq[col], s2);
  }
}

// ---------------------------------------------------------------- BatchNorm
__global__ void bn_finalize_kernel(const float* __restrict__ colsum,
                                   const float* __restrict__ colsumsq,
                                   const float* __restrict__ gamma,
                                   const float* __restrict__ beta,
                                   float* __restrict__ bnA, float* __restrict__ bnB) {
  int c = threadIdx.x;
  if (c >= DD) return;
  const float inv = 1.0f / (float)NN;
  float mu = colsum[c] * inv;
  float var = colsumsq[c] * inv - mu * mu;
  float a = gamma[c] * rsqrtf(var + BN_EPS);
  bnA[c] = a;
  bnB[c] = fmaf(-mu, a, beta[c]);
}

__global__ void bn_apply_relu_kernel(const float* __restrict__ h,
                                     const float* __restrict__ bnA,
                                     const float* __restrict__ bnB,
                                     float* __restrict__ xout) {
  int i = blockIdx.x * blockDim.x + threadIdx.x;
  if (i >= NN * DD / 4) return;
  int base = i << 2;
  int c = base & (DD - 1);
  float4 v = *(const float4*)(h + base);
  const float4 A = *(const float4*)(bnA + c);
  const float4 B = *(const float4*)(bnB + c);
  v.x = fmaxf(fmaf(v.x, A.x, B.x), 0.f);
  v.y = fmaxf(fmaf(v.y, A.y, B.y), 0.f);
  v.z = fmaxf(fmaf(v.z, A.z, B.z), 0.f);
  v.w = fmaxf(fmaf(v.w, A.w, B.w), 0.f);
  *(float4*)(xout + base) = v;
}

// -------------------------------------------------------------- final head
// out[n] = dot(h1[n, 0:64], Wc2) + bc2 ; one wave per node, 2 floats per lane.
__global__ void head_kernel(const float* __restrict__ h1, const float* __restrict__ w2,
                            const float* __restrict__ b2, float* __restrict__ out) {
  int t = blockIdx.x * blockDim.x + threadIdx.x;
  int node = t >> 5;
  if (node >= NN) return;
  int lane = t & 31;
  const float2 v = *(const float2*)(h1 + (size_t)node * HCOLS + lane * 2);
  const float2 w = *(const float2*)(w2 + lane * 2);
  float s = fmaf(v.x, w.x, v.y * w.y);
#pragma unroll
  for (int off = 16; off > 0; off >>= 1) s += __shfl_xor(s, off, 32);
  if (lane == 0) out[node] = s + b2[0];
}

// ---------------------------------------------------------------- launcher
extern "C" void kernel_launch(void* const* d_in, const int* in_sizes, int n_in,
                              void* d_out, int out_size, void* d_ws, size_t ws_size,
                              hipStream_t stream) {
  const float* x = (const float*)d_in[0];
  const int* ei = (const int*)d_in[1];
  const int* src = ei;
  const int* dst = ei + NE;
  const float* Wl[3] = {(const float*)d_in[2], (const float*)d_in[7], (const float*)d_in[12]};
  const float* bl[3] = {(const float*)d_in[3], (const float*)d_in[8], (const float*)d_in[13]};
  const float* Wr[3] = {(const float*)d_in[4], (const float*)d_in[9], (const float*)d_in[14]};
  const float* gm[3] = {(const float*)d_in[5], (const float*)d_in[10], (const float*)d_in[15]};
  const float* be[3] = {(const float*)d_in[6], (const float*)d_in[11], (const float*)d_in[16]};
  const float* Wc1 = (const float*)d_in[17];
  const float* bc1 = (const float*)d_in[18];
  const float* Wc2 = (const float*)d_in[19];
  const float* bc2 = (const float*)d_in[20];
  float* out = (float*)d_out;

  // Workspace layout (256B aligned). h and the classifier hidden alias `agg`.
  char* ws = (char*)d_ws;
  size_t off = 0;
  auto take = [&](size_t bytes) -> float* {
    float* p = (float*)(ws + off);
    off += (bytes + 255) & ~(size_t)255;
    return p;
  };
  float* deg      = take((size_t)NN * 4);            // becomes deg_inv
  float* agg      = take((size_t)NN * DD * 4);       // agg / h / classifier hidden
  float* xA       = take((size_t)NN * DD * 4);
  float* xB       = take((size_t)NN * DD * 4);
  float* colsum   = take(DD * 4);
  float* colsumsq = take(DD * 4);
  float* bnA      = take(DD * 4);
  float* bnB      = take(DD * 4);
  (void)ws_size; (void)n_in; (void)in_sizes; (void)out_size;

  const int B = 256;

  // degree -> deg_inv (recomputed every call; deterministic)
  zero4_kernel<<<(NN / 4 + B - 1) / B, B, 0, stream>>>((float4*)deg, NN / 4);
  degree_kernel<<<(NE + B - 1) / B, B, 0, stream>>>(dst, deg);
  deginv_kernel<<<(NN + B - 1) / B, B, 0, stream>>>(deg);

  const float* xin = x;
  for (int l = 0; l < 3; ++l) {
    zero4_kernel<<<(NN * DD / 4 + B - 1) / B, B, 0, stream>>>((float4*)agg, NN * DD / 4);
    zero4_kernel<<<1, 64, 0, stream>>>((float4*)colsum, DD / 4);
    zero4_kernel<<<1, 64, 0, stream>>>((float4*)colsumsq, DD / 4);
    aggregate_kernel<<<(NE * 32) / B, B, 0, stream>>>(xin, src, dst, agg);
    // h = (agg * deg_inv) @ Wl^T + bl + xin @ Wr^T, with BN column stats
    wmma_gemm_kernel<true, false, true><<<NN / 16, 256, 0, stream>>>(
        agg, deg, xin, Wl[l], Wr[l], bl[l], agg, DD, colsum, colsumsq);
    bn_finalize_kernel<<<1, DD, 0, stream>>>(colsum, colsumsq, gm[l], be[l], bnA, bnB);
    float* xout = (l & 1) ? xB : xA;  // l0->xA, l1->xB, l2->xA
    bn_apply_relu_kernel<<<(NN * DD / 4 + B - 1) / B, B, 0, stream>>>(agg, bnA, bnB, xout);
    xin = xout;
  }

  // classifier: h1 = relu(xin @ Wc1^T + bc1)  [N,64] (4 waves -> 64 cols)
  wmma_gemm_kernel<false, true, false><<<NN / 16, 128, 0, stream>>>(
      xin, nullptr, nullptr, Wc1, nullptr, bc1, agg, HCOLS, nullptr, nullptr);
  // out = h1 @ Wc2^T + bc2
  head_kernel<<<(NN * 32 + B - 1) / B, B, 0, stream>>>(agg, Wc2, bc2, out);
}